// WeightedGaussianPool_32375463477475
// MI455X (gfx1250) — compile-verified
//
#include <hip/hip_runtime.h>
#include <math.h>

// CDNA5 / gfx1250. wave32. WMMA f32 16x16x4 for the pairwise Gram tiles,
// branchless f32 VALU + v_exp_f32/v_cos_f32 pipeline for the Gaussian basis.

typedef __attribute__((ext_vector_type(2))) float v2f;
typedef __attribute__((ext_vector_type(4))) float v4f;
typedef __attribute__((ext_vector_type(8))) float v8f;

#define MAXN   2048
#define NB     64
#define NWAVES 16
#define BLOCK  (NWAVES * 32)

__launch_bounds__(BLOCK)
__global__ void WeightedGaussianPool_kernel(const float* __restrict__ f,
                                            const float* __restrict__ coords,
                                            const float* __restrict__ means,
                                            const float* __restrict__ betaPtr,
                                            float* __restrict__ out, int N)
{
    // 64KB LDS, reused:
    //   stage:  [0 .. 2N)   interleaved (x,y) pairs        (K=0,1 fragment halves)
    //           [2N .. 4N)  interleaved (z,0) pairs        (K=2,3 fragment halves)
    //           [4N .. 5N)  q = |c|^2
    //           [5N .. 6N)  f
    //   reduce: 16 waves x 16 n x 64 b partials
    __shared__ float s[NWAVES * 16 * NB];   // 16384 floats = 65536 B
    float* sXY = s;                          // float2[i] = (x,y)
    float* sZ0 = s + 2 * MAXN;               // float2[i] = (z,0)
    float* sQ  = s + 4 * MAXN;
    float* sF  = s + 5 * MAXN;               // ends at 12288 <= 16384

    const int tid = threadIdx.x;
    const float invCut = 0.2f;               // 1 / CUTOFF(=5)

    // Stage (pad with f=0 -> zero weight for padded m's).
    for (int i = tid; i < MAXN; i += BLOCK) {
        float x = 0.f, y = 0.f, z = 0.f, q = 0.f, fv = 0.f;
        if (i < N) {
            x = coords[3 * i + 0] * invCut;
            y = coords[3 * i + 1] * invCut;
            z = coords[3 * i + 2] * invCut;
            q = x * x + y * y + z * z;
            fv = f[i];
        }
        sXY[2 * i + 0] = x; sXY[2 * i + 1] = y;
        sZ0[2 * i + 0] = z; sZ0[2 * i + 1] = 0.f;
        sQ[i] = q; sF[i] = fv;
    }
    __syncthreads();

    const int wave = tid >> 5;
    const int lane = tid & 31;
    const int l16  = lane & 15;
    const int hi   = lane >> 4;              // 0: K=0,1 (x,y)   1: K=2,3 (z,0)

    const float beta = betaPtr[0];
    const float negc = -beta * 1.44269504088896f;  // exp(-b t^2) = exp2(negc * t^2)

    const int nTile = blockIdx.x;
    const int np = nTile * 16 + l16;

    // Fragments live as (x,y) or (z,0) float2 in LDS -> one ds_load_b64, no branch.
    const v2f* fragBase = (const v2f*)(hi ? sZ0 : sXY);   // uniform per half-wave
    const v2f bfrag = fragBase[np];
    const float cn2 = sQ[np];

    float acc[NB];
#pragma unroll
    for (int b = 0; b < NB; ++b) acc[b] = 0.0f;

    const int nTilesM = MAXN / 16;           // 128; padded tiles contribute 0 (f=0)
    for (int mt = wave; mt < nTilesM; mt += NWAVES) {
        const int mBase = mt * 16;
        const v2f afrag = fragBase[mBase + l16];

        v8f czero = {0.f, 0.f, 0.f, 0.f, 0.f, 0.f, 0.f, 0.f};
        // D[m_row, n_col] tile of dot(c_m, c_n): v_wmma_f32_16x16x4_f32
        v8f dots = __builtin_amdgcn_wmma_f32_16x16x4_f32(
            false, afrag, false, bfrag, (short)0, czero, false, false);

        const int mSub = mBase + 8 * hi;     // this lane's 8 m rows (16B aligned)
        const v4f q0 = *(const v4f*)(sQ + mSub);
        const v4f q1 = *(const v4f*)(sQ + mSub + 4);
        const v4f f0 = *(const v4f*)(sF + mSub);
        const v4f f1 = *(const v4f*)(sF + mSub + 4);

#pragma unroll
        for (int v = 0; v < 8; ++v) {
            const float qm = (v < 4) ? q0[v & 3] : q1[v & 3];
            const float fm = (v < 4) ? f0[v & 3] : f1[v & 3];
            float sq = cn2 + qm - 2.0f * dots[v];
            sq = fmaxf(sq, 1e-12f);
            const float d = __builtin_amdgcn_sqrtf(sq);
            // cos(pi/2 * d) = v_cos_f32(d/4)  (hw cos takes revolutions)
            float w = fm * __builtin_amdgcn_cosf(0.25f * d);
            w = (d <= 1.0f) ? w : 0.0f;      // branchless step -> v_cndmask
#pragma unroll
            for (int b = 0; b < NB; ++b) {   // trans-bound core: 1 v_exp_f32 per term
                const float t = d - means[b];            // means[] -> SGPRs
                const float e = __builtin_amdgcn_exp2f((negc * t) * t);
                acc[b] = fmaf(w, e, acc[b]);
            }
        }
    }

    // Fold lane <-> lane+16 (same n, complementary m rows).
#pragma unroll
    for (int b = 0; b < NB; ++b) acc[b] += __shfl_xor(acc[b], 16, 32);

    __syncthreads();                         // staging data dead; reuse LDS
    if (hi == 0) {
        float* row = s + (wave * 16 + l16) * NB;
#pragma unroll
        for (int b = 0; b < NB; b += 4)
            *(v4f*)(row + b) = (v4f){acc[b], acc[b + 1], acc[b + 2], acc[b + 3]};
    }
    __syncthreads();

    // Deterministic cross-wave tree: 1024 outputs, 512 threads x 2 each.
    for (int r = tid; r < 16 * NB; r += BLOCK) {
        const int n = r >> 6, b = r & 63;
        float sum = 0.f;
#pragma unroll
        for (int w2 = 0; w2 < NWAVES; ++w2) sum += s[(w2 * 16 + n) * NB + b];
        const int gn = nTile * 16 + n;
        if (gn < N) out[gn * NB + b] = sum;
    }
}

extern "C" void kernel_launch(void* const* d_in, const int* in_sizes, int n_in,
                              void* d_out, int out_size, void* d_ws, size_t ws_size,
                              hipStream_t stream) {
    const float* f      = (const float*)d_in[0];   // [N,1] float32
    const float* coords = (const float*)d_in[1];   // [N,3] float32
    const float* means  = (const float*)d_in[2];   // [64]  float32
    const float* beta   = (const float*)d_in[3];   // [1]   float32
    float* out = (float*)d_out;                    // [N,64] float32

    const int N = in_sizes[0];
    const int grid = (N + 15) / 16;
    WeightedGaussianPool_kernel<<<grid, BLOCK, 0, stream>>>(f, coords, means, beta, out, N);
}